// mv_embedding_37039797961005
// MI455X (gfx1250) — compile-verified
//
#include <hip/hip_runtime.h>

// Problem constants (match reference)
#define B_DIM   512
#define K_DIM   512
#define D_IN    1024
#define D_K     1536          // 3*K
#define D_OUT   4096
#define ROWS    (B_DIM * K_DIM)   // 262144 (b,k) rows
#define NBLK_A  1024
#define WAVES_A (NBLK_A * 8)
#define CHUNK   512
#define EPSV    1e-5f

typedef float v2f __attribute__((ext_vector_type(2)));
typedef float v8f __attribute__((ext_vector_type(8)));

// ---------------------------------------------------------------------------
// Kernel A: y[b,k,c] = dot(x[b,k,:], W_emb[c,:]) + b_emb[c]
// Streams 1 GiB of x (HBM-bound phase). W_emb slice held in registers.
// Also produces per-block partial sums / sums-of-squares per channel
// (deterministic: wave shfl reduce + LDS tree, no float atomics).
// Y stored in the reshape layout: Y[b*1536 + c*512 + k].
// ---------------------------------------------------------------------------
__global__ __launch_bounds__(256) void embed_stats_kernel(
    const float* __restrict__ x, const float* __restrict__ W_emb,
    const float* __restrict__ b_emb, float* __restrict__ Y,
    float* __restrict__ partials)
{
    const int tid  = threadIdx.x;
    const int lane = tid & 31;
    const int wave = tid >> 5;
    const int gw   = blockIdx.x * 8 + wave;

    // Per-lane register slice of W_emb: lane covers j = t*128 + lane*4 .. +3
    float4 w0[8], w1[8], w2[8];
#pragma unroll
    for (int t = 0; t < 8; ++t) {
        const int j = t * 128 + lane * 4;
        w0[t] = *(const float4*)(W_emb + 0 * D_IN + j);
        w1[t] = *(const float4*)(W_emb + 1 * D_IN + j);
        w2[t] = *(const float4*)(W_emb + 2 * D_IN + j);
    }
    const float be0 = b_emb[0], be1 = b_emb[1], be2 = b_emb[2];

    float s0 = 0.f, s1 = 0.f, s2 = 0.f;
    float q0 = 0.f, q1 = 0.f, q2 = 0.f;

    for (int row = gw; row < ROWS; row += WAVES_A) {
        const float* xr = x + (size_t)row * D_IN;
        float a0 = 0.f, a1 = 0.f, a2 = 0.f;
#pragma unroll
        for (int t = 0; t < 8; ++t) {
            const float4 xv = *(const float4*)(xr + t * 128 + lane * 4);
            a0 = fmaf(xv.x, w0[t].x, a0); a0 = fmaf(xv.y, w0[t].y, a0);
            a0 = fmaf(xv.z, w0[t].z, a0); a0 = fmaf(xv.w, w0[t].w, a0);
            a1 = fmaf(xv.x, w1[t].x, a1); a1 = fmaf(xv.y, w1[t].y, a1);
            a1 = fmaf(xv.z, w1[t].z, a1); a1 = fmaf(xv.w, w1[t].w, a1);
            a2 = fmaf(xv.x, w2[t].x, a2); a2 = fmaf(xv.y, w2[t].y, a2);
            a2 = fmaf(xv.z, w2[t].z, a2); a2 = fmaf(xv.w, w2[t].w, a2);
        }
        // wave32 butterfly reduction (deterministic order)
#pragma unroll
        for (int off = 16; off > 0; off >>= 1) {
            a0 += __shfl_xor(a0, off, 32);
            a1 += __shfl_xor(a1, off, 32);
            a2 += __shfl_xor(a2, off, 32);
        }
        a0 += be0; a1 += be1; a2 += be2;
        if (lane == 0) {
            const int b = row >> 9;          // /512
            const int k = row & 511;
            float* yr = Y + (size_t)b * D_K + k;
            yr[0]    = a0;
            yr[512]  = a1;
            yr[1024] = a2;
            s0 += a0; q0 = fmaf(a0, a0, q0);
            s1 += a1; q1 = fmaf(a1, a1, q1);
            s2 += a2; q2 = fmaf(a2, a2, q2);
        }
    }

    __shared__ float red[8][6];
    if (lane == 0) {
        red[wave][0] = s0; red[wave][1] = s1; red[wave][2] = s2;
        red[wave][3] = q0; red[wave][4] = q1; red[wave][5] = q2;
    }
    __syncthreads();
    if (tid < 6) {
        float acc = 0.f;
#pragma unroll
        for (int w = 0; w < 8; ++w) acc += red[w][tid];
        partials[blockIdx.x * 6 + tid] = acc;   // [0..2]=sum, [3..5]=sumsq
    }
}

// ---------------------------------------------------------------------------
// Kernel B: reduce 1024 partials -> per-channel scale/shift.
// stats[c] = gamma[c]*rsqrt(var+eps) ; stats[3+c] = beta[c] - mean*scale
// ---------------------------------------------------------------------------
__global__ void finalize_stats_kernel(
    const float* __restrict__ partials, const float* __restrict__ gamma,
    const float* __restrict__ beta, float* __restrict__ stats)
{
    const int lane = threadIdx.x;   // 32 threads, 1 wave
    float acc[6];
#pragma unroll
    for (int c = 0; c < 6; ++c) {
        float a = 0.f;
        for (int i = lane; i < NBLK_A; i += 32) a += partials[i * 6 + c];
#pragma unroll
        for (int off = 16; off > 0; off >>= 1) a += __shfl_xor(a, off, 32);
        acc[c] = a;
    }
    if (lane == 0) {
        const float invN = 1.0f / (float)ROWS;
#pragma unroll
        for (int c = 0; c < 3; ++c) {
            const float mean = acc[c] * invN;
            const float var  = fmaf(-mean, mean, acc[c + 3] * invN);
            const float sc   = gamma[c] * rsqrtf(var + EPSV);
            stats[c]     = sc;
            stats[3 + c] = fmaf(-mean, sc, beta[c]);
        }
    }
}

// ---------------------------------------------------------------------------
// Kernel C: out[512,4096] = BN(Y)[512,1536] @ W_fc2^T, fp32 WMMA 16x16x4.
// Block: 256 threads / 8 waves -> 16 rows x 512 cols of output.
// A-chunk (16x512, one BN channel) staged in 32 KB LDS with BN affine folded
// into the load. Each wave owns 4 16x16 accumulator tiles (A-frag reuse 4x).
//
// Fragment layouts per CDNA5 ISA 7.12.2 (wave32, half = lane>>4, m = lane&15):
//   A 16x4 f32 : lane holds A[m][2*half + {0,1}]        (v2f)
//   B 4x16 f32 : lane holds B[2*half + {0,1}][m]        (v2f)
//   C/D 16x16  : acc[r] = D[r + 8*half][m]              (v8f)
// ---------------------------------------------------------------------------
__global__ __launch_bounds__(256) void gemm2_wmma_kernel(
    const float* __restrict__ Y, const float* __restrict__ W_fc2,
    const float* __restrict__ stats, float* __restrict__ out)
{
    __shared__ float As[16 * CHUNK];   // 32 KB

    const int tid     = threadIdx.x;
    const int lane    = tid & 31;
    const int wave    = tid >> 5;
    const int half    = lane >> 4;
    const int m       = lane & 15;
    const int row0    = blockIdx.y * 16;
    const int colbase = blockIdx.x * 512 + wave * 64;

    v8f acc[4] = {};

    for (int kc = 0; kc < D_K; kc += CHUNK) {
        const int   c  = kc >> 9;         // chunk is channel-uniform
        const float sc = stats[c];
        const float sh = stats[3 + c];

        __syncthreads();
        // stage normalized A chunk: 16 rows x 512 cols = 2048 float4
#pragma unroll
        for (int r = 0; r < 8; ++r) {
            const int flat = tid + r * 256;        // float4 index
            const int arow = flat >> 7;            // /128 float4-per-row
            const int acol = (flat & 127) << 2;
            float4 v = *(const float4*)(Y + (size_t)(row0 + arow) * D_K + kc + acol);
            v.x = fmaf(v.x, sc, sh); v.y = fmaf(v.y, sc, sh);
            v.z = fmaf(v.z, sc, sh); v.w = fmaf(v.w, sc, sh);
            *(float4*)(As + arow * CHUNK + acol) = v;
        }
        __syncthreads();

        for (int k0 = 0; k0 < CHUNK; k0 += 4) {
            const v2f a = *(const v2f*)(As + m * CHUNK + k0 + 2 * half);
#pragma unroll
            for (int t = 0; t < 4; ++t) {
                const v2f b = *(const v2f*)(W_fc2 +
                    (size_t)(colbase + t * 16 + m) * D_K + kc + k0 + 2 * half);
                acc[t] = __builtin_amdgcn_wmma_f32_16x16x4_f32(
                    false, a, false, b, (short)0, acc[t], false, false);
            }
        }
    }

#pragma unroll
    for (int t = 0; t < 4; ++t) {
#pragma unroll
        for (int r = 0; r < 8; ++r) {
            out[(size_t)(row0 + r + 8 * half) * D_OUT + colbase + t * 16 + m] =
                acc[t][r];
        }
    }
}

// ---------------------------------------------------------------------------
extern "C" void kernel_launch(void* const* d_in, const int* in_sizes, int n_in,
                              void* d_out, int out_size, void* d_ws, size_t ws_size,
                              hipStream_t stream) {
    const float* x      = (const float*)d_in[0];   // [512,512,1024]
    const float* W_emb  = (const float*)d_in[1];   // [3,1024]
    const float* b_emb  = (const float*)d_in[2];   // [3]
    const float* gamma  = (const float*)d_in[3];   // [3]
    const float* beta   = (const float*)d_in[4];   // [3]
    const float* W_fc2  = (const float*)d_in[5];   // [4096,1536]
    float*       out    = (float*)d_out;           // [512,4096]

    float* ws       = (float*)d_ws;
    float* Y        = ws;                          // 512*1536 floats (3 MB)
    float* partials = Y + (size_t)B_DIM * D_K;     // 1024*6 floats
    float* stats    = partials + NBLK_A * 6;       // 6 floats

    embed_stats_kernel<<<NBLK_A, 256, 0, stream>>>(x, W_emb, b_emb, Y, partials);
    finalize_stats_kernel<<<1, 32, 0, stream>>>(partials, gamma, beta, stats);
    gemm2_wmma_kernel<<<dim3(8, 32), 256, 0, stream>>>(Y, W_fc2, stats, out);
}